// ImputationLoss_10514079941417
// MI455X (gfx1250) — compile-verified
//
#include <hip/hip_runtime.h>
#include <hip/hip_bf16.h>

#define BDIM 32
#define LDIM 4096
#define CDIM 4        // classes; label == CDIM means "missing"
#define CF   5        // y_true channels
#define NG   8        // B / GROUP_SIZE
#define CHUNK 128     // L-chunk per WMMA workgroup
#define JS_EPSF 1e-7f

typedef float v2f __attribute__((ext_vector_type(2)));
typedef float v8f __attribute__((ext_vector_type(8)));

__device__ __forceinline__ float block_reduce_256(float v, float* s) {
    int t = threadIdx.x;
    s[t] = v;
    __syncthreads();
    for (int off = 128; off > 0; off >>= 1) {
        if (t < off) s[t] += s[t + off];
        __syncthreads();
    }
    float r = s[0];
    __syncthreads();
    return r;
}

// ---------------- Phase 1: per-(b,l) labels, CE, plogp ----------------
__global__ __launch_bounds__(256)
void prep_kernel(const float* __restrict__ logits,
                 const float* __restrict__ prob,
                 const float* __restrict__ y_true,
                 float* __restrict__ plogp_out,
                 int*   __restrict__ lbl_out,
                 float* __restrict__ scal) {
    __shared__ float sred[256];
    int idx = blockIdx.x * blockDim.x + threadIdx.x;   // b*L + l
    float ce = 0.f, val = 0.f;
    if (idx < BDIM * LDIM) {
        const float* yt = y_true + (size_t)idx * CF;
        float best = yt[0]; int lbl = 0;
#pragma unroll
        for (int c = 1; c < CF; ++c) {
            float v = yt[c];
            if (v > best) { best = v; lbl = c; }   // first-max tie rule (argmax)
        }
        lbl_out[idx] = lbl;

        float4 p4 = ((const float4*)prob)[idx];
        float p0 = fmaxf(p4.x, JS_EPSF), p1 = fmaxf(p4.y, JS_EPSF);
        float p2 = fmaxf(p4.z, JS_EPSF), p3 = fmaxf(p4.w, JS_EPSF);
        plogp_out[idx] = p0 * __logf(p0) + p1 * __logf(p1) +
                         p2 * __logf(p2) + p3 * __logf(p3);

        if (lbl != CDIM) {
            float4 lg = ((const float4*)logits)[idx];
            float lx0 = lg.x, lx1 = lg.y, lx2 = lg.z, lx3 = lg.w;
            float m = fmaxf(fmaxf(lx0, lx1), fmaxf(lx2, lx3));
            float s = __expf(lx0 - m) + __expf(lx1 - m) +
                      __expf(lx2 - m) + __expf(lx3 - m);
            float xl = (lbl == 0) ? lx0 : (lbl == 1) ? lx1 : (lbl == 2) ? lx2 : lx3;
            ce = -(xl - m - __logf(s));
            val = 1.f;
        }
    }
    float ce_t  = block_reduce_256(ce,  sred);
    float val_t = block_reduce_256(val, sred);
    if (threadIdx.x == 0) {
        atomicAdd(&scal[0], ce_t);
        atomicAdd(&scal[1], val_t);
    }
}

// ---------------- Phase 2: minimac R2 over groups of 4 ----------------
__global__ __launch_bounds__(256)
void r2_kernel(const float* __restrict__ prob,
               const float* __restrict__ y_true,
               const int*   __restrict__ lbl,
               float* __restrict__ scal) {
    __shared__ float sred[256];
    int idx = blockIdx.x * blockDim.x + threadIdx.x;   // g*L + l
    float lr2 = 0.f, ltv = 0.f;
    if (idx < NG * LDIM) {
        int g = idx / LDIM, l = idx - g * LDIM;
        float nm_cnt = 0.f, alt = 0.f, pred = 0.f;
        int lb[4];
#pragma unroll
        for (int k = 0; k < 4; ++k) {
            size_t bl = (size_t)(g * 4 + k) * LDIM + l;
            lb[k] = lbl[bl];
            nm_cnt += (lb[k] != CDIM) ? 1.f : 0.f;
            const float* yt = y_true + bl * CF;
            alt += yt[1] + yt[2] + yt[3];
            float4 p = ((const float4*)prob)[bl];
            pred += p.y + p.z + p.w;
        }
        pred *= 0.25f;
        if (nm_cnt > 0.f) {
            float af = alt / fmaxf(nm_cnt, 1.f);
            if (af != 0.f && af != 1.f) {            // edge check
                float denom = fmaxf(af * (1.f - af), 0.01f);
                float d = pred - af;
                float r2 = d * d / denom;
                if (r2 != 0.f) {                     // gvm: group_r2 != 0
                    lr2 = -r2 * nm_cnt;              // one term per nonmissing member
                    ltv = nm_cnt;
                }
            }
        }
    }
    float r2_t = block_reduce_256(lr2, sred);
    float tv_t = block_reduce_256(ltv, sred);
    if (threadIdx.x == 0) {
        atomicAdd(&scal[2], r2_t);
        atomicAdd(&scal[3], tv_t);
    }
}

// ---------------- Phase 3a: T1 = A*B^T via V_WMMA_F32_16X16X4_F32 ----------------
// A(i, k=(l,c)) = plogp(i,l) * [lbl(i,l)==c] ; B(j, k) = [lbl(j,l)==c]
// One K-step of 4 covers one l with c=0..3.
// A frag (16x4 f32, 2 VGPRs): lane t -> M=t&15, K={c0,c0+1}, c0 = (t>=16)?2:0
// B frag (4x16 f32, 2 VGPRs): lane t -> N=t&15, same K mapping (assumed symmetric layout)
// D frag (16x16 f32, 8 VGPRs): VGPR v -> M = v + (t<16 ? 0 : 8), N = t&15
__global__ __launch_bounds__(32)
void evo_wmma_kernel(const float* __restrict__ plogp,
                     const int*   __restrict__ lbl,
                     float* __restrict__ T1) {
    __shared__ float s_pp[32][CHUNK];
    __shared__ int   s_lb[32][CHUNK];
    int t  = threadIdx.x;            // single wave32, EXEC all-ones for WMMA
    int l0 = blockIdx.x * CHUNK;
    for (int l = 0; l < CHUNK; ++l) {
        s_pp[t][l] = plogp[(size_t)t * LDIM + l0 + l];
        s_lb[t][l] = lbl  [(size_t)t * LDIM + l0 + l];
    }
    __syncthreads();

    int m  = t & 15;
    int c0 = (t >> 4) << 1;          // 0 or 2
    v8f acc00 = {0.f,0.f,0.f,0.f,0.f,0.f,0.f,0.f};
    v8f acc01 = acc00, acc10 = acc00, acc11 = acc00;

    for (int l = 0; l < CHUNK; ++l) {
        int   li0 = s_lb[m][l];      float pp0 = s_pp[m][l];
        int   li1 = s_lb[16 + m][l]; float pp1 = s_pp[16 + m][l];
        v2f a0, a1, b0, b1;
        a0.x = (li0 == c0    ) ? pp0 : 0.f;
        a0.y = (li0 == c0 + 1) ? pp0 : 0.f;
        a1.x = (li1 == c0    ) ? pp1 : 0.f;
        a1.y = (li1 == c0 + 1) ? pp1 : 0.f;
        b0.x = (li0 == c0    ) ? 1.f : 0.f;
        b0.y = (li0 == c0 + 1) ? 1.f : 0.f;
        b1.x = (li1 == c0    ) ? 1.f : 0.f;
        b1.y = (li1 == c0 + 1) ? 1.f : 0.f;
        acc00 = __builtin_amdgcn_wmma_f32_16x16x4_f32(false, a0, false, b0, (short)0, acc00, false, false);
        acc01 = __builtin_amdgcn_wmma_f32_16x16x4_f32(false, a0, false, b1, (short)0, acc01, false, false);
        acc10 = __builtin_amdgcn_wmma_f32_16x16x4_f32(false, a1, false, b0, (short)0, acc10, false, false);
        acc11 = __builtin_amdgcn_wmma_f32_16x16x4_f32(false, a1, false, b1, (short)0, acc11, false, false);
    }

    int Mbase = (t < 16) ? 0 : 8;
#pragma unroll
    for (int v = 0; v < 8; ++v) {
        int Mi = Mbase + v;
        atomicAdd(&T1[(Mi     ) * 32 + (m     )], acc00[v]);
        atomicAdd(&T1[(Mi     ) * 32 + (m + 16)], acc01[v]);
        atomicAdd(&T1[(Mi + 16) * 32 + (m     )], acc10[v]);
        atomicAdd(&T1[(Mi + 16) * 32 + (m + 16)], acc11[v]);
    }
}

// ---------------- Phase 3b: Msum(i,j) = sum_l mask * (m log m) ----------------
__global__ __launch_bounds__(256)
void pairwise_kernel(const float* __restrict__ prob,
                     const int*   __restrict__ lbl,
                     float* __restrict__ Msum) {
    __shared__ float sred[256];
    int pair = blockIdx.x;             // 1024 pairs
    int i = pair >> 5, j = pair & 31;
    int t = threadIdx.x;
    const float4* p4 = (const float4*)prob;
    float acc = 0.f;
    for (int l = t; l < LDIM; l += 256) {
        int li = lbl[(size_t)i * LDIM + l];
        int lj = lbl[(size_t)j * LDIM + l];
        if (li == lj && li != CDIM) {
            float4 pi = p4[(size_t)i * LDIM + l];
            float4 pj = p4[(size_t)j * LDIM + l];
            float m0 = 0.5f * (fmaxf(pi.x, JS_EPSF) + fmaxf(pj.x, JS_EPSF));
            float m1 = 0.5f * (fmaxf(pi.y, JS_EPSF) + fmaxf(pj.y, JS_EPSF));
            float m2 = 0.5f * (fmaxf(pi.z, JS_EPSF) + fmaxf(pj.z, JS_EPSF));
            float m3 = 0.5f * (fmaxf(pi.w, JS_EPSF) + fmaxf(pj.w, JS_EPSF));
            acc += m0 * __logf(m0) + m1 * __logf(m1) +
                   m2 * __logf(m2) + m3 * __logf(m3);
        }
    }
    float total = block_reduce_256(acc, sred);
    if (t == 0) Msum[pair] = total;
}

// ---------------- Phase 4: combine everything ----------------
__global__ __launch_bounds__(32)
void final_kernel(const float* __restrict__ y_evo,
                  const float* __restrict__ T1,
                  const float* __restrict__ Msum,
                  const float* __restrict__ scal,
                  float* __restrict__ out) {
    int i = threadIdx.x;   // 0..31, one wave
    float wsum = 0.f;
    for (int j = 0; j < 32; ++j) wsum += __expf(-2.f * y_evo[i * 32 + j]);
    float contrib = 0.f;
    for (int j = 0; j < 32; ++j) {
        float w  = __expf(-2.f * y_evo[i * 32 + j]) / (wsum + 1e-8f);
        float js = (i == j) ? 0.f
                 : (0.5f * (T1[i * 32 + j] + T1[j * 32 + i]) - Msum[i * 32 + j]);
        contrib += w * js;
    }
    for (int off = 16; off > 0; off >>= 1)
        contrib += __shfl_down(contrib, off, 32);
    if (i == 0) {
        float evo = 10.f * (contrib / 32.f);
        if (!isfinite(evo)) evo = 0.f;
        float ce_sum = scal[0], valid = scal[1], r2_sum = scal[2], tv = scal[3];
        float ce = (valid > 0.f) ? (ce_sum / fmaxf(valid, 1.f)) : ce_sum;
        float r2 = 0.1f * ((tv > 0.f) ? (r2_sum / fmaxf(tv, 1.f)) : r2_sum);
        out[0] = 1000.f * (ce + r2 + evo);
    }
}

extern "C" void kernel_launch(void* const* d_in, const int* in_sizes, int n_in,
                              void* d_out, int out_size, void* d_ws, size_t ws_size,
                              hipStream_t stream) {
    (void)in_sizes; (void)n_in; (void)out_size; (void)ws_size;
    const float* logits = (const float*)d_in[0];
    const float* prob   = (const float*)d_in[1];
    const float* y_true = (const float*)d_in[2];
    const float* y_evo  = (const float*)d_in[3];
    float* out = (float*)d_out;

    float* ws    = (float*)d_ws;
    float* T1    = ws;                       // 1024 floats (32x32)
    float* scal  = ws + 1024;                // [ce_sum, valid, r2_sum, tv]
    float* Msum  = ws + 2048;                // 1024 floats (32x32)
    float* plogp = ws + 4096;                // B*L floats
    int*   lbl   = (int*)(ws + 4096 + BDIM * LDIM);  // B*L ints

    // zero T1 + scalars every call (deterministic re-run; graph-capturable memset)
    hipMemsetAsync(ws, 0, 2048 * sizeof(float), stream);

    prep_kernel<<<(BDIM * LDIM + 255) / 256, 256, 0, stream>>>(
        logits, prob, y_true, plogp, lbl, scal);
    r2_kernel<<<(NG * LDIM + 255) / 256, 256, 0, stream>>>(
        prob, y_true, lbl, scal);
    evo_wmma_kernel<<<LDIM / CHUNK, 32, 0, stream>>>(plogp, lbl, T1);
    pairwise_kernel<<<1024, 256, 0, stream>>>(prob, lbl, Msum);
    final_kernel<<<1, 32, 0, stream>>>(y_evo, T1, Msum, scal, out);
}